// ISC_constraint_and_IC_loss_22677427323529
// MI455X (gfx1250) — compile-verified
//
#include <hip/hip_runtime.h>
#include <hip/hip_bf16.h>
#include <math.h>

// ---------------------------------------------------------------------------
// LCD loss (IC + ISC) + sim_batch for B=512, C=128 on gfx1250 (MI455X).
//
// S[a,b] = sum_c p[a,c]*log(p[a,c]/(p[b,c]+eps)+eps)
//        = D[a] + (P * Lneg^T)[a,b] + (Reps * P^T)[a,b]        (exact to <1e-10)
// where D[a]=sum_c p*log(p+eps^2), Lneg=-log(p+eps), Reps=eps*p/(p+eps^2).
// The two B x B x C f32 GEMMs run on V_WMMA_F32_16X16X4_F32 with the masked
// |S| reduction fused in the epilogue. ISC needs only 2 exact terms per pair.
// 4 independent accumulators break the WMMA RAW chain (1 wave/SIMD => in-wave
// ILP is the only latency hider on this tiny, launch-latency-bound problem).
// ---------------------------------------------------------------------------

#define Bsz 512
#define Csz 128
#define NTILE 32            // 512/16
#define NPAIR_TILES 528     // tiles with tb >= ta
#define LDST 132            // LDS row stride (floats): 16B aligned, bank-skewed
#define EPS 1e-6f
#define EPS2 1e-12f

typedef float v2f __attribute__((ext_vector_type(2)));
typedef float v8f __attribute__((ext_vector_type(8)));

__device__ __forceinline__ float waveSum(float v) {
  v += __shfl_xor(v, 16, 32);
  v += __shfl_xor(v, 8, 32);
  v += __shfl_xor(v, 4, 32);
  v += __shfl_xor(v, 2, 32);
  v += __shfl_xor(v, 1, 32);
  return v;
}

__device__ __forceinline__ bool betterOf(float va, int ia, float vb, int ib) {
  return (va > vb) || (va == vb && ia < ib);   // argsort stability: lower idx
}

// ---------------- kernel 0: zero sim_batch region of d_out -----------------
__global__ __launch_bounds__(256) void zeroKernel(float* simB) {
  simB[blockIdx.x * 256 + threadIdx.x] = 0.0f;
}

// ------- kernel 1: per-row softmax, derived matrices, D, top-2 scatter -----
__global__ __launch_bounds__(256) void rowKernel(
    const float* __restrict__ pred, const int* __restrict__ labels,
    const float* __restrict__ Tp, float* __restrict__ P,
    float* __restrict__ Ln, float* __restrict__ Re,
    float* __restrict__ Dv, float* __restrict__ simB) {
  const int lane = threadIdx.x & 31;
  const int wid  = threadIdx.x >> 5;
  const int a    = blockIdx.x * 8 + wid;        // one wave per row
  const float T  = Tp[0];
  const float* row = pred + a * Csz;

  float x[4];
#pragma unroll
  for (int j = 0; j < 4; ++j) x[j] = row[lane + 32 * j] / T;

  float m = fmaxf(fmaxf(x[0], x[1]), fmaxf(x[2], x[3]));
  m = fmaxf(m, __shfl_xor(m, 16, 32));
  m = fmaxf(m, __shfl_xor(m, 8, 32));
  m = fmaxf(m, __shfl_xor(m, 4, 32));
  m = fmaxf(m, __shfl_xor(m, 2, 32));
  m = fmaxf(m, __shfl_xor(m, 1, 32));

  float e[4], s = 0.0f;
#pragma unroll
  for (int j = 0; j < 4; ++j) { e[j] = expf(x[j] - m); s += e[j]; }
  s = waveSum(s);

  float d = 0.0f;
  float p[4];
#pragma unroll
  for (int j = 0; j < 4; ++j) {
    p[j] = e[j] / s;
    const int c = lane + 32 * j;
    P[a * Csz + c]  = p[j];
    Ln[a * Csz + c] = -logf(p[j] + EPS);
    Re[a * Csz + c] = EPS * (p[j] / (p[j] + EPS2));
    d += p[j] * logf(p[j] + EPS2);
  }
  d = waveSum(d);

  // local top-2 (value, index), ties -> lower index
  float v1 = -INFINITY, v2 = -INFINITY;
  int i1 = -1, i2 = -1;
#pragma unroll
  for (int j = 0; j < 4; ++j) {
    const int c = lane + 32 * j;
    if (betterOf(p[j], c, v1, i1)) { v2 = v1; i2 = i1; v1 = p[j]; i1 = c; }
    else if (betterOf(p[j], c, v2, i2)) { v2 = p[j]; i2 = c; }
  }
#pragma unroll
  for (int off = 16; off >= 1; off >>= 1) {
    float ov1 = __shfl_xor(v1, off, 32), ov2 = __shfl_xor(v2, off, 32);
    int   oi1 = __shfl_xor(i1, off, 32), oi2 = __shfl_xor(i2, off, 32);
    if (betterOf(ov1, oi1, v1, i1)) { v2 = v1; i2 = i1; v1 = ov1; i1 = oi1; }
    else if (betterOf(ov1, oi1, v2, i2)) { v2 = ov1; i2 = oi1; }
    if (betterOf(ov2, oi2, v1, i1)) { v2 = v1; i2 = i1; v1 = ov2; i1 = oi2; }
    else if (betterOf(ov2, oi2, v2, i2)) { v2 = ov2; i2 = oi2; }
  }

  if (lane == 0) {
    Dv[a] = d;
    const int lab = labels[a];
    if (lab == i1) atomicAdd(&simB[lab * Csz + i2], 1.0f);  // exact int adds
  }
}

// ---------- kernel 2: WMMA f32 tiles, fused masked |S| reduction -----------
__device__ __forceinline__ void copyTile(float* dst, const float* src) {
  for (int i = threadIdx.x; i < 512; i += 32) {       // 16 rows x 32 float4
    const int r = i >> 5, c = i & 31;
    const float4 v = reinterpret_cast<const float4*>(src)[r * 32 + c];
    *reinterpret_cast<float4*>(dst + r * LDST + c * 4) = v;
  }
}

__global__ __launch_bounds__(32) void tileKernel(
    const float* __restrict__ P, const float* __restrict__ Ln,
    const float* __restrict__ Re, const float* __restrict__ Dv,
    const int* __restrict__ labels, float* __restrict__ icPart) {
  __shared__ float sPa[16 * LDST], sLb[16 * LDST], sRa[16 * LDST], sPb[16 * LDST];

  // decode triangular tile (ta, tb) with tb >= ta  (scalar, uniform)
  int rem = blockIdx.x, ta = 0;
  while (rem >= NTILE - ta) { rem -= NTILE - ta; ++ta; }
  const int tb = ta + rem;
  const int a0 = ta * 16, b0 = tb * 16;

  copyTile(sPa, P + a0 * Csz);
  copyTile(sLb, Ln + b0 * Csz);
  copyTile(sRa, Re + a0 * Csz);
  copyTile(sPb, P + b0 * Csz);
  // single wave: DS ops are in-order within a wave; no barrier needed

  const int lane = threadIdx.x;
  const int half = lane >> 4;          // K sub-block select (ISA A/B layouts)
  const int lid  = lane & 15;          // A row M / B column N
  const float* pa = sPa + lid * LDST;
  const float* lb = sLb + lid * LDST;
  const float* ra = sRa + lid * LDST;
  const float* pb = sPb + lid * LDST;

  // 4 independent accumulator chains (2 GEMMs x even/odd k-step)
  v8f acc0 = {0.f, 0.f, 0.f, 0.f, 0.f, 0.f, 0.f, 0.f};
  v8f acc1 = acc0, acc2 = acc0, acc3 = acc0;
#pragma unroll 4
  for (int k = 0; k < Csz; k += 8) {
    const int k0 = k + 2 * half;
    const int k1 = k0 + 4;
    v2f aP0 = *(const v2f*)(pa + k0);
    v2f bL0 = *(const v2f*)(lb + k0);
    v2f aR0 = *(const v2f*)(ra + k0);
    v2f bP0 = *(const v2f*)(pb + k0);
    v2f aP1 = *(const v2f*)(pa + k1);
    v2f bL1 = *(const v2f*)(lb + k1);
    v2f aR1 = *(const v2f*)(ra + k1);
    v2f bP1 = *(const v2f*)(pb + k1);
    acc0 = __builtin_amdgcn_wmma_f32_16x16x4_f32(false, aP0, false, bL0,
                                                 (short)0, acc0, false, false);
    acc1 = __builtin_amdgcn_wmma_f32_16x16x4_f32(false, aR0, false, bP0,
                                                 (short)0, acc1, false, false);
    acc2 = __builtin_amdgcn_wmma_f32_16x16x4_f32(false, aP1, false, bL1,
                                                 (short)0, acc2, false, false);
    acc3 = __builtin_amdgcn_wmma_f32_16x16x4_f32(false, aR1, false, bP1,
                                                 (short)0, acc3, false, false);
  }

  // epilogue: C/D layout VGPR v -> M = v + 8*half, N = lid (branchless mask)
  float local = 0.0f;
  const int bi = b0 + lid;
  const int labB = labels[bi];
#pragma unroll
  for (int v = 0; v < 8; ++v) {
    const int ai = a0 + v + 8 * half;
    const float S = Dv[ai] + ((acc0[v] + acc1[v]) + (acc2[v] + acc3[v]));
    const float msk = (bi > ai && labels[ai] == labB) ? 1.0f : 0.0f;
    local += msk * fabsf(S);
  }
  local = waveSum(local);
  if (lane == 0) icPart[blockIdx.x] = local;   // deterministic partial
}

// --------- kernel 3: exact ISC terms (2 per pair) + pair counts ------------
__global__ __launch_bounds__(256) void pairKernel(
    const float* __restrict__ P, const int* __restrict__ labels,
    const float* __restrict__ simAll, const int* __restrict__ epochP,
    float* __restrict__ iscPart, float* __restrict__ scPart,
    float* __restrict__ dcPart) {
  __shared__ float red[3][8];
  const int idx = blockIdx.x * 256 + threadIdx.x;
  const int a = idx >> 9, b = idx & (Bsz - 1);
  float isc = 0.0f, sc = 0.0f, dc = 0.0f;
  if (b > a) {
    const int la = labels[a], lb = labels[b];
    if (la == lb) {
      sc = 1.0f;
    } else {
      dc = 1.0f;
      const float paa = P[a * Csz + la], pba = P[b * Csz + la];
      const float pab = P[a * Csz + lb], pbb = P[b * Csz + lb];
      const float t1 = paa * logf(paa / (pba + EPS) + EPS);
      const float t2 = pab * logf(pab / (pbb + EPS) + EPS);
      const float sim = (epochP[0] == 0) ? 1.0f : simAll[la * Csz + lb];
      isc = fabsf((t1 + t2) * sim);
    }
  }
  isc = waveSum(isc); sc = waveSum(sc); dc = waveSum(dc);
  const int lane = threadIdx.x & 31, wid = threadIdx.x >> 5;
  if (lane == 0) { red[0][wid] = isc; red[1][wid] = sc; red[2][wid] = dc; }
  __syncthreads();
  if (threadIdx.x == 0) {
    float s0 = 0.f, s1 = 0.f, s2 = 0.f;
    for (int i = 0; i < 8; ++i) { s0 += red[0][i]; s1 += red[1][i]; s2 += red[2][i]; }
    iscPart[blockIdx.x] = s0; scPart[blockIdx.x] = s1; dcPart[blockIdx.x] = s2;
  }
}

// --------------- kernel 4: deterministic final reduction -------------------
__device__ __forceinline__ float blockReduceArr(const float* arr, int n,
                                                float* buf) {
  float s = 0.0f;
  for (int i = threadIdx.x; i < n; i += 256) s += arr[i];  // fixed order
  buf[threadIdx.x] = s;
  __syncthreads();
  for (int off = 128; off > 0; off >>= 1) {
    if (threadIdx.x < off) buf[threadIdx.x] += buf[threadIdx.x + off];
    __syncthreads();
  }
  const float r = buf[0];
  __syncthreads();
  return r;
}

__global__ __launch_bounds__(256) void finKernel(
    const float* __restrict__ icPart, const float* __restrict__ iscPart,
    const float* __restrict__ scPart, const float* __restrict__ dcPart,
    const float* __restrict__ muP, const float* __restrict__ etaP,
    float* __restrict__ out) {
  __shared__ float buf[256];
  float IC  = blockReduceArr(icPart, NPAIR_TILES, buf);
  float ISC = blockReduceArr(iscPart, 1024, buf);
  float SC  = blockReduceArr(scPart, 1024, buf);
  float DC  = blockReduceArr(dcPart, 1024, buf);
  if (threadIdx.x == 0) {
    if (SC != 0.0f) IC = IC / SC;
    if (DC != 0.0f) ISC = ISC / DC;
    if (ISC != 0.0f) ISC = (1.0f / (ISC + EPS)) * muP[0];
    IC = IC * etaP[0];
    out[0] = IC + ISC;
  }
}

// ---------------------------------------------------------------------------
extern "C" void kernel_launch(void* const* d_in, const int* in_sizes, int n_in,
                              void* d_out, int out_size, void* d_ws, size_t ws_size,
                              hipStream_t stream) {
  (void)in_sizes; (void)n_in; (void)out_size; (void)ws_size;
  const float* predicts = (const float*)d_in[0];
  const int*   labels   = (const int*)d_in[1];
  const float* sim_all  = (const float*)d_in[2];
  const int*   epoch    = (const int*)d_in[3];
  const float* T        = (const float*)d_in[4];
  const float* mu       = (const float*)d_in[5];
  const float* eta      = (const float*)d_in[6];

  float* out  = (float*)d_out;
  float* simB = out + 1;                         // (C,C) sim_batch

  float* ws = (float*)d_ws;                      // ~785 KB of f32 scratch
  float* P  = ws;                                // 512*128
  float* Ln = ws + 65536;                        // 512*128
  float* Re = ws + 131072;                       // 512*128
  float* Dv = ws + 196608;                       // 512
  float* icPart  = ws + 197120;                  // 528
  float* iscPart = icPart + NPAIR_TILES;         // 1024
  float* scPart  = iscPart + 1024;               // 1024
  float* dcPart  = scPart + 1024;                // 1024

  zeroKernel<<<64, 256, 0, stream>>>(simB);
  rowKernel<<<64, 256, 0, stream>>>(predicts, labels, T, P, Ln, Re, Dv, simB);
  tileKernel<<<NPAIR_TILES, 32, 0, stream>>>(P, Ln, Re, Dv, labels, icPart);
  pairKernel<<<1024, 256, 0, stream>>>(P, labels, sim_all, epoch,
                                       iscPart, scPart, dcPart);
  finKernel<<<1, 256, 0, stream>>>(icPart, iscPart, scPart, dcPart, mu, eta, out);
}